// VectorMapOffroadLoss_50105088475726
// MI455X (gfx1250) — compile-verified
//
#include <hip/hip_runtime.h>
#include <hip/hip_bf16.h>
#include <stdint.h>

typedef __attribute__((ext_vector_type(2))) float v2f;
typedef __attribute__((ext_vector_type(8))) float v8f;

#define BB 4
#define NN 64
#define TT 80
#define SS 1024
#define NPTS (NN * TT)                 // 5120 points per batch
#define TILES_PER_B (NPTS / 16)        // 320 tiles of 16 points
#define WAVES_PER_BLK 8
#define BLKS_PER_B (TILES_PER_B / WAVES_PER_BLK)  // 40
#define NFEAT 9

// feature array indices (each SS floats, per batch)
#define F_ABX  0
#define F_ABY  1
#define F_C0   2   // -(a . ab)
#define F_M2AX 3   // -2*ax
#define F_M2AY 4   // -2*ay
#define F_AN   5   // ||a||^2
#define F_INV  6   // 1/len_sq
#define F_LSQ  7   // len_sq
#define F_VB   8   // 0 if valid else 1e18 (added to d^2)

// workspace layout (float offsets)
#define WS_FEAT 0
#define WS_WTAB (BB * NFEAT * SS)       // 36864: 80 decay weights
#define WS_INAV (WS_WTAB + TT)          // 36944: 4 x 1/max(sum av,1)
#define WS_HSEG (WS_INAV + BB)          // 36948: 4 x has_seg flag
#define WS_PART (WS_HSEG + BB)          // 36952: 4*320 per-tile partials

// ---------------------------------------------------------------------------
// Kernel 1: decay-weight table (serial, reproducible), per-batch agent count
// inverse and has_seg flag. Tiny; one block.
// ---------------------------------------------------------------------------
__global__ void misc_kernel(const uint8_t* __restrict__ seg_valid,
                            const uint8_t* __restrict__ agent_valid,
                            float* __restrict__ ws) {
  int tid = threadIdx.x;
  if (tid == 0) {
    float tmp[TT];
    float w = 1.0f, s = 0.0f;
    for (int t = 0; t < TT; ++t) { tmp[t] = w; s += w; w *= 0.9f; }
    float inv = 1.0f / s;
    for (int t = 0; t < TT; ++t) ws[WS_WTAB + t] = tmp[t] * inv;
  }
  if (tid >= 32 && tid < 32 + BB) {
    int b = tid - 32;
    float cnt = 0.0f;
    for (int n = 0; n < NN; ++n) cnt += agent_valid[b * NN + n] ? 1.0f : 0.0f;
    ws[WS_INAV + b] = 1.0f / fmaxf(cnt, 1.0f);
  }
  if (tid >= 64 && tid < 64 + BB) {
    int b = tid - 64;
    int any = 0;
    for (int s = 0; s < SS; ++s) any |= (int)seg_valid[b * SS + s];
    ws[WS_HSEG + b] = any ? 1.0f : 0.0f;
  }
}

// ---------------------------------------------------------------------------
// Kernel 2: per-segment WMMA B-matrix features.
// proj  = [px,py,pn,1] . [abx, aby, 0,  -a.ab ]
// ||ap||^2 = [px,py,pn,1] . [-2ax,-2ay,1, ||a||^2]   (pn = ||p||^2)
// ---------------------------------------------------------------------------
__global__ void feat_kernel(const float* __restrict__ seg_a,
                            const float* __restrict__ seg_b,
                            const uint8_t* __restrict__ seg_valid,
                            float* __restrict__ ws) {
  int idx = blockIdx.x * blockDim.x + threadIdx.x;   // b*SS + s
  if (idx >= BB * SS) return;
  int b = idx / SS, s = idx - b * SS;
  float ax = seg_a[idx * 2 + 0], ay = seg_a[idx * 2 + 1];
  float bx = seg_b[idx * 2 + 0], by = seg_b[idx * 2 + 1];
  float abx = bx - ax, aby = by - ay;
  float lsq = fmaxf(abx * abx + aby * aby, 1e-6f);
  float* f = ws + WS_FEAT + (size_t)b * NFEAT * SS;
  f[F_ABX  * SS + s] = abx;
  f[F_ABY  * SS + s] = aby;
  f[F_C0   * SS + s] = -(ax * abx + ay * aby);
  f[F_M2AX * SS + s] = -2.0f * ax;
  f[F_M2AY * SS + s] = -2.0f * ay;
  f[F_AN   * SS + s] = ax * ax + ay * ay;
  f[F_INV  * SS + s] = 1.0f / lsq;
  f[F_LSQ  * SS + s] = lsq;
  f[F_VB   * SS + s] = seg_valid[idx] ? 0.0f : 1e18f;
}

// ---------------------------------------------------------------------------
// Kernel 3: main WMMA kernel. Block = 8 waves; each wave owns one 16-point
// tile; all waves share the batch's segment features staged in LDS (36 KB).
// Per segment tile (16 segs): 2x V_WMMA_F32_16X16X4_F32 -> proj & ||ap||^2,
// then clamp/dist/min in VALU. Cross-lane min via ds_swizzle-class shuffles.
// ---------------------------------------------------------------------------
__global__ __launch_bounds__(256) void
offroad_main_kernel(const float* __restrict__ Y,
                    const uint8_t* __restrict__ agent_valid,
                    const float* __restrict__ ws,
                    float* __restrict__ partials) {
  __shared__ float sm[NFEAT * SS];

  const int b   = blockIdx.y;
  const int tid = threadIdx.x;
  const float* feat = ws + WS_FEAT + (size_t)b * NFEAT * SS;
  for (int i = tid; i < NFEAT * SS; i += 256) sm[i] = feat[i];
  __syncthreads();

  const int wave = tid >> 5;
  const int lane = tid & 31;
  const int half = lane >> 4;       // 0: K=0..1 rows, 1: K=2..3 rows
  const int col  = lane & 15;       // M for A-frag, N for B/C frags
  const int tile = blockIdx.x * WAVES_PER_BLK + wave;

  // A fragment: 16x4, lane M = col, VGPR pair holds K = 2*half .. 2*half+1
  int P = tile * 16 + col;
  int n = P / TT, t = P - n * TT;
  const float* yp = Y + (((size_t)(b * NN + n) * TT + t) * 4);
  float px = yp[0], py = yp[1];
  v2f afrag;
  if (half == 0) { afrag.x = px;            afrag.y = py;   }
  else           { afrag.x = px * px + py * py; afrag.y = 1.0f; }

  float mind2[8];
#pragma unroll
  for (int r = 0; r < 8; ++r) mind2[r] = 3.0e38f;

  for (int st = 0; st < SS / 16; ++st) {
    const int s = st * 16 + col;
    // B fragments: 4x16, lane N = col, VGPR pair holds K = 2*half .. 2*half+1
    v2f bproj, bpa2;
    if (half == 0) {
      bproj.x = sm[F_ABX  * SS + s]; bproj.y = sm[F_ABY  * SS + s];
      bpa2.x  = sm[F_M2AX * SS + s]; bpa2.y  = sm[F_M2AY * SS + s];
    } else {
      bproj.x = 0.0f;                bproj.y = sm[F_C0 * SS + s];
      bpa2.x  = 1.0f;                bpa2.y  = sm[F_AN * SS + s];
    }
    v8f dproj = {};
    v8f dpa2  = {};
    dproj = __builtin_amdgcn_wmma_f32_16x16x4_f32(
        false, afrag, false, bproj, (short)0, dproj, false, false);
    dpa2  = __builtin_amdgcn_wmma_f32_16x16x4_f32(
        false, afrag, false, bpa2,  (short)0, dpa2,  false, false);

    const float inv = sm[F_INV * SS + s];
    const float lsq = sm[F_LSQ * SS + s];
    const float vb  = sm[F_VB  * SS + s];
#pragma unroll
    for (int r = 0; r < 8; ++r) {
      float pr = dproj[r];
      float tc = fminf(fmaxf(pr * inv, 0.0f), 1.0f);
      float u  = __builtin_fmaf(-tc, lsq, pr + pr);   // 2*proj - t*lsq
      float d2 = __builtin_fmaf(-tc, u, dpa2[r]);     // ||ap||^2 - t*(...)
      d2 = fmaxf(d2, 0.0f) + vb;                      // invalid seg -> +1e18
      mind2[r] = fminf(mind2[r], d2);
    }
  }

  // min over the 16 segment columns of this lane-half (M = r + 8*half)
#pragma unroll
  for (int r = 0; r < 8; ++r) {
#pragma unroll
    for (int off = 1; off < 16; off <<= 1)
      mind2[r] = fminf(mind2[r], __shfl_xor(mind2[r], off, 32));
  }

  // per-half weighted offroad^2 sum for its 8 rows
  float wsum = 0.0f;
#pragma unroll
  for (int r = 0; r < 8; ++r) {
    int M  = r + 8 * half;
    int Pm = tile * 16 + M;
    int nm = Pm / TT, tm = Pm - nm * TT;
    float md  = sqrtf(mind2[r]);
    float ofr = fmaxf(md - 2.0f, 0.0f);
    float av  = agent_valid[b * NN + nm] ? 1.0f : 0.0f;
    wsum += ofr * ofr * ws[WS_WTAB + tm] * av;
  }
  float tot = wsum + __shfl_xor(wsum, 16, 32);  // combine the two halves
  if (lane == 0) partials[b * TILES_PER_B + tile] = tot;
}

// ---------------------------------------------------------------------------
// Kernel 4: deterministic final reduction -> scalar.
// ---------------------------------------------------------------------------
__global__ void reduce_kernel(const float* __restrict__ ws,
                              const float* __restrict__ partials,
                              float* __restrict__ out) {
  __shared__ float sb[BB];
  int tid = threadIdx.x;
  int wave = tid >> 5, lane = tid & 31;
  if (wave < BB) {
    int b = wave;
    float s = 0.0f;
    for (int i = lane; i < TILES_PER_B; i += 32)
      s += partials[b * TILES_PER_B + i];
    for (int off = 16; off >= 1; off >>= 1) s += __shfl_xor(s, off, 32);
    if (lane == 0) sb[b] = s * ws[WS_INAV + b] * ws[WS_HSEG + b];
  }
  __syncthreads();
  if (tid == 0)
    out[0] = (sb[0] + sb[1] + sb[2] + sb[3]) * (1.0f / (float)BB);
}

// ---------------------------------------------------------------------------
extern "C" void kernel_launch(void* const* d_in, const int* in_sizes, int n_in,
                              void* d_out, int out_size, void* d_ws,
                              size_t ws_size, hipStream_t stream) {
  (void)in_sizes; (void)n_in; (void)out_size; (void)ws_size;
  const float*   Y           = (const float*)d_in[0];
  const float*   seg_a       = (const float*)d_in[1];
  const float*   seg_b       = (const float*)d_in[2];
  const uint8_t* seg_valid   = (const uint8_t*)d_in[3];  // numpy bool = 1 byte
  const uint8_t* agent_valid = (const uint8_t*)d_in[4];
  float* ws  = (float*)d_ws;
  float* out = (float*)d_out;
  float* partials = ws + WS_PART;

  misc_kernel<<<1, 128, 0, stream>>>(seg_valid, agent_valid, ws);
  feat_kernel<<<(BB * SS + 255) / 256, 256, 0, stream>>>(seg_a, seg_b,
                                                         seg_valid, ws);
  dim3 grid(BLKS_PER_B, BB);
  offroad_main_kernel<<<grid, 256, 0, stream>>>(Y, agent_valid, ws, partials);
  reduce_kernel<<<1, 128, 0, stream>>>(ws, partials, out);
}